// MixProp_Dlinear_53154515256297
// MI455X (gfx1250) — compile-verified
//
#include <hip/hip_runtime.h>
#include <hip/hip_bf16.h>

typedef __attribute__((ext_vector_type(2))) float v2f;
typedef __attribute__((ext_vector_type(8))) float v8f;

#define NB   2000   // N nodes
#define TT   336    // T
#define BB   4      // batch
#define HH   96     // H
#define BT   (BB*TT)        // 1344
#define MT   (NB/16)        // 125 m-tiles
#define PROP_A 0.05f

static __device__ __forceinline__ v8f wmma_f32(v2f a, v2f b, v8f c) {
    return __builtin_amdgcn_wmma_f32_16x16x4_f32(false, a, false, b, (short)0, c,
                                                 false, false);
}

// ---------------- prep: collapse all small linear layers to beta[3], bias[96] ---
__global__ __launch_bounds__(128)
void prep_scalars(const float* __restrict__ start_w, const float* __restrict__ start_b,
                  const float* __restrict__ mlp_w,   const float* __restrict__ mlp_b,
                  const float* __restrict__ seas_b,  const float* __restrict__ trend_w,
                  const float* __restrict__ trend_b, const float* __restrict__ f1_w,
                  const float* __restrict__ f1_b,    const float* __restrict__ f2_w,
                  const float* __restrict__ f2_b,
                  float* __restrict__ betas, float* __restrict__ bias)
{
    __shared__ float sK, sc0, sg;
    if (threadIdx.x == 0) {
        float g[32];
        for (int c = 0; c < 32; ++c) {
            float s = 0.f;
            for (int j = 0; j < 6; ++j) s += f2_w[j] * f1_w[j * 32 + c];
            g[c] = s;
        }
        float Kc = f2_b[0];
        for (int j = 0; j < 6; ++j) Kc += f2_w[j] * f1_b[j];
        float gsum = 0.f;
        for (int c = 0; c < 32; ++c) gsum += g[c];
        float u[3] = {0.f, 0.f, 0.f};
        float c0 = 0.f;
        for (int o = 0; o < 32; ++o) {
            float Co = mlp_b[o];
            float Wo[3] = {0.f, 0.f, 0.f};
            for (int k = 0; k < 3; ++k)
                for (int c = 0; c < 32; ++c) {
                    float mw = mlp_w[o * 96 + k * 32 + c];
                    Wo[k] += mw * start_w[c];
                    Co    += mw * start_b[c];
                }
            for (int k = 0; k < 3; ++k) u[k] += g[o] * Wo[k];
            c0 += g[o] * Co;
        }
        const float al = PROP_A, om = 1.f - PROP_A;
        betas[0] = u[0] + al * (u[1] + u[2]);
        betas[1] = om * (u[1] + al * u[2]);
        betas[2] = om * om * u[2];
        sK = Kc; sc0 = c0; sg = gsum;
    }
    __syncthreads();
    int h = threadIdx.x;
    if (h < HH) {
        float es = 0.f;                       // sum_t E[h,t] == sum_t trend_w[h,t]
        for (int t = 0; t < TT; ++t) es += trend_w[h * TT + t];
        bias[h] = sK + sc0 * es + sg * (seas_b[h] + trend_b[h]);
    }
}

// ---------------- Et[t,h]: effective DLinear time projection (transposed) -------
__global__ __launch_bounds__(256)
void build_Et(const float* __restrict__ seas_w, const float* __restrict__ trend_w,
              float* __restrict__ Et)
{
    int idx = blockIdx.x * 256 + threadIdx.x;
    if (idx >= HH * TT) return;
    int h = idx / TT, t = idx % TT;
    const float* sw = seas_w + h * TT;
    const float* tw = trend_w + h * TT;
    float acc = 0.f;
    int lo = (t - 3 < 0) ? 0 : t - 3;
    int hi = (t + 3 > TT - 1) ? TT - 1 : t + 3;
    for (int tp = lo; tp <= hi; ++tp) acc += tw[tp] - sw[tp];
    if (t == 0)
        acc += 3.f * (tw[0] - sw[0]) + 2.f * (tw[1] - sw[1]) + (tw[2] - sw[2]);
    if (t == TT - 1)
        acc += 3.f * (tw[TT-1] - sw[TT-1]) + 2.f * (tw[TT-2] - sw[TT-2]) + (tw[TT-3] - sw[TT-3]);
    Et[t * HH + h] = sw[t] + acc * (1.f / 7.f);
}

// ---------------- row-normalize (adj + I), store transposed aT[w,v] -------------
__global__ __launch_bounds__(256)
void rownorm_T(const float* __restrict__ adj, float* __restrict__ aT)
{
    int v = blockIdx.x;
    __shared__ float red[256];
    float s = 0.f;
    for (int w = threadIdx.x; w < NB; w += 256)
        s += adj[(size_t)v * NB + w] + (w == v ? 1.f : 0.f);
    red[threadIdx.x] = s;
    __syncthreads();
    for (int off = 128; off > 0; off >>= 1) {
        if (threadIdx.x < off) red[threadIdx.x] += red[threadIdx.x + off];
        __syncthreads();
    }
    float inv = 1.f / red[0];
    for (int w = threadIdx.x; w < NB; w += 256) {
        float val = adj[(size_t)v * NB + w] + (w == v ? 1.f : 0.f);
        aT[(size_t)w * NB + v] = val * inv;
    }
}

// ---------------- Sg[n, b*T+t] = batch_x[b,t,n]  (flat X idx == j*N+n) ----------
__global__ __launch_bounds__(256)
void build_S(const float* __restrict__ X, float* __restrict__ Sg)
{
    int idx = blockIdx.x * 256 + threadIdx.x;
    if (idx >= NB * BT) return;
    int n = idx % NB;
    int j = idx / NB;
    Sg[(size_t)n * BT + j] = X[idx];
}

// ---------------- C[M,J] = A * B, A given transposed aT[k,m], all f32 WMMA ------
// wave computes 16(M) x 64(J); block = 8 waves = 8 m-tiles at one j64 tile.
__global__ __launch_bounds__(256)
void gemm_at(const float* __restrict__ aT, const float* __restrict__ B,
             float* __restrict__ C, int K, int ldb, int ldc)
{
    const int lane  = threadIdx.x & 31;
    const int wave  = threadIdx.x >> 5;
    const int mt    = blockIdx.x * 8 + wave;
    if (mt >= MT) return;
    const int m0    = mt << 4;
    const int j0    = blockIdx.y * 64;
    const int lh    = lane & 15;
    const int khi   = (lane >> 4) << 1;          // 0 or 2
    v8f acc0 = {}, acc1 = {}, acc2 = {}, acc3 = {};
    for (int k = 0; k < K; k += 4) {
        const int ka = k + khi;
        v2f a;
        a.x = aT[(size_t)ka * NB + m0 + lh];
        a.y = aT[(size_t)(ka + 1) * NB + m0 + lh];
        const float* bp0 = B + (size_t)ka * ldb + j0 + lh;
        const float* bp1 = B + (size_t)(ka + 1) * ldb + j0 + lh;
        v2f b0; b0.x = bp0[0];  b0.y = bp1[0];
        v2f b1; b1.x = bp0[16]; b1.y = bp1[16];
        v2f b2; b2.x = bp0[32]; b2.y = bp1[32];
        v2f b3; b3.x = bp0[48]; b3.y = bp1[48];
        acc0 = wmma_f32(a, b0, acc0);
        acc1 = wmma_f32(a, b1, acc1);
        acc2 = wmma_f32(a, b2, acc2);
        acc3 = wmma_f32(a, b3, acc3);
    }
    const int hi8 = (lane >> 4) << 3;            // 0 or 8
#pragma unroll
    for (int r = 0; r < 8; ++r) {
        float* cp = C + (size_t)(m0 + r + hi8) * ldc + j0 + lh;
        cp[0]  = acc0[r];
        cp[16] = acc1[r];
        cp[32] = acc2[r];
        cp[48] = acc3[r];
    }
}

// ---------------- Z = b0*S + b1*H1 + b2*H2   (in-place over Sg) -----------------
__global__ __launch_bounds__(256)
void combine_Z(float* __restrict__ Sg, const float* __restrict__ H1,
               const float* __restrict__ H2, const float* __restrict__ betas)
{
    int idx = blockIdx.x * 256 + threadIdx.x;
    if (idx >= NB * BT) return;
    Sg[idx] = betas[0] * Sg[idx] + betas[1] * H1[idx] + betas[2] * H2[idx];
}

// ---------------- out[b,h,n] = sum_t Z[n, b*T+t] * Et[t,h] + bias[h] ------------
// wave: 16 n-rows x 96 h (6 WMMA frags); 500 waves total.
__global__ __launch_bounds__(256)
void final_proj(const float* __restrict__ Z, const float* __restrict__ Et,
                const float* __restrict__ bias, float* __restrict__ out)
{
    const int lane = threadIdx.x & 31;
    const int wave = threadIdx.x >> 5;
    int gid = blockIdx.x * 8 + wave;
    if (gid >= BB * MT) return;
    int b  = gid / MT;
    int m0 = (gid % MT) << 4;
    const int lh  = lane & 15;
    const int khi = (lane >> 4) << 1;
    v8f acc[6];
#pragma unroll
    for (int f = 0; f < 6; ++f) acc[f] = (v8f){};
    const float* Zb = Z + (size_t)b * TT;
    for (int k = 0; k < TT; k += 4) {
        const int ka = k + khi;
        const float* ap = Zb + (size_t)(m0 + lh) * BT + ka;
        v2f a; a.x = ap[0]; a.y = ap[1];
#pragma unroll
        for (int f = 0; f < 6; ++f) {
            v2f bb;
            bb.x = Et[(size_t)ka * HH + f * 16 + lh];
            bb.y = Et[(size_t)(ka + 1) * HH + f * 16 + lh];
            acc[f] = wmma_f32(a, bb, acc[f]);
        }
    }
    const int hi8 = (lane >> 4) << 3;
#pragma unroll
    for (int f = 0; f < 6; ++f) {
        int h = f * 16 + lh;
        float bv = bias[h];
#pragma unroll
        for (int r = 0; r < 8; ++r) {
            int n = m0 + r + hi8;
            out[(size_t)b * HH * NB + (size_t)h * NB + n] = acc[f][r] + bv;
        }
    }
}

extern "C" void kernel_launch(void* const* d_in, const int* in_sizes, int n_in,
                              void* d_out, int out_size, void* d_ws, size_t ws_size,
                              hipStream_t stream)
{
    const float* batch_x = (const float*)d_in[0];
    const float* adj     = (const float*)d_in[1];
    const float* start_w = (const float*)d_in[2];
    const float* start_b = (const float*)d_in[3];
    const float* mlp_w   = (const float*)d_in[4];
    const float* mlp_b   = (const float*)d_in[5];
    const float* seas_w  = (const float*)d_in[6];
    const float* seas_b  = (const float*)d_in[7];
    const float* trend_w = (const float*)d_in[8];
    const float* trend_b = (const float*)d_in[9];
    const float* f1_w    = (const float*)d_in[10];
    const float* f1_b    = (const float*)d_in[11];
    const float* f2_w    = (const float*)d_in[12];
    const float* f2_b    = (const float*)d_in[13];
    float* out = (float*)d_out;

    float* ws    = (float*)d_ws;
    float* aT    = ws;                        // 4,000,000 f
    float* Sg    = aT + (size_t)NB * NB;      // 2,688,000 f (becomes Z in place)
    float* H1    = Sg + (size_t)NB * BT;      // 2,688,000 f
    float* H2    = H1 + (size_t)NB * BT;      // 2,688,000 f
    float* Et    = H2 + (size_t)NB * BT;      // 32,256 f
    float* bias  = Et + TT * HH;              // 96 f
    float* betas = bias + HH;                 // 3 f   (total ~48.4 MB)

    prep_scalars<<<1, 128, 0, stream>>>(start_w, start_b, mlp_w, mlp_b, seas_b,
                                        trend_w, trend_b, f1_w, f1_b, f2_w, f2_b,
                                        betas, bias);
    build_Et<<<(HH * TT + 255) / 256, 256, 0, stream>>>(seas_w, trend_w, Et);
    rownorm_T<<<NB, 256, 0, stream>>>(adj, aT);
    build_S<<<(NB * BT + 255) / 256, 256, 0, stream>>>(batch_x, Sg);

    dim3 g1((MT + 7) / 8, BT / 64);           // 16 x 21
    gemm_at<<<g1, 256, 0, stream>>>(aT, Sg, H1, NB, BT, BT);   // H1 = a @ S
    gemm_at<<<g1, 256, 0, stream>>>(aT, H1, H2, NB, BT, BT);   // H2 = a @ H1

    combine_Z<<<(NB * BT + 255) / 256, 256, 0, stream>>>(Sg, H1, H2, betas);

    final_proj<<<(BB * MT + 7) / 8, 256, 0, stream>>>(Sg, Et, bias, out);
}